// GCNConvLayer_85203561218535
// MI455X (gfx1250) — compile-verified
//
#include <hip/hip_runtime.h>

typedef float v2f __attribute__((ext_vector_type(2)));
typedef float v4f __attribute__((ext_vector_type(4)));
typedef float v8f __attribute__((ext_vector_type(8)));
typedef unsigned int u32x4 __attribute__((ext_vector_type(4)));
typedef int i32x8 __attribute__((ext_vector_type(8)));

#define D_DIM 128
#define EPSBN 1e-5f
#define STAT_BLOCKS 512
#define KCHUNK 64
#define LDS_ROWSTRIDE 66   /* 64 data DWORDs + 2 pad DWORDs -> 8B aligned, no bank conflicts */

// ---------------------------------------------------------------------------
// TDM: issue a 2D tile load Global -> LDS.
//   D# group0: count=1 | lds_addr | global_addr | type=2
//   D# group1: data_size=4B, pad 2 DWORDs every 64 DWORDs, dims/strides
// Caller must be a single wave (TDM ignores EXEC; one issue per wave).
// ---------------------------------------------------------------------------
__device__ __forceinline__ void tdm_load_tile_2d(
    unsigned ldsByteAddr, const void* gaddr,
    int tensorRows, int strideElems, int tileCols, int tileRows)
{
  unsigned long long ga = (unsigned long long)(uintptr_t)gaddr;
  u32x4 g0;
  g0[0] = 1u;                                      // count=1, user mode
  g0[1] = ldsByteAddr;                             // lds_addr
  g0[2] = (unsigned)(ga & 0xFFFFFFFFu);            // global_addr[31:0]
  g0[3] = (unsigned)((ga >> 32) & 0x1FFFFFFu)      // global_addr[56:32]
          | (2u << 30);                            // type = 2 (image)
  i32x8 g1;
  g1[0] = (2 << 16)      // data_size = 4 bytes
        | (1 << 20)      // pad_enable
        | (5 << 22)      // pad_interval: every 64 DWORDs
        | (1 << 25);     // pad_amount: 2 DWORDs
  g1[1] = (tileCols  & 0xFFFF) << 16;              // tensor_dim0[15:0]
  g1[2] = (tensorRows & 0xFFFF) << 16;             // tensor_dim1[15:0] (dim0 hi16 = 0)
  g1[3] = (tileCols  & 0xFFFF) << 16;              // tile_dim0 (dim1 hi16 = 0)
  g1[4] = (tileRows  & 0xFFFF);                    // tile_dim1 (tile_dim2 = 0)
  g1[5] = strideElems;                             // tensor_dim0_stride[31:0]
  g1[6] = 0;
  g1[7] = 0;
  asm volatile("tensor_load_to_lds %0, %1" :: "s"(g0), "s"(g1) : "memory");
}

// ---------------------------------------------------------------------------
// GEMM: C[M,OUT] = A[M,K] @ Bw[OUT,K]^T  (+ optional bias / relu / residual)
// fp32 WMMA 16x16x4. Block = 8 waves -> 16 rows x 128 cols of C.
// Per K-chunk: TDM stages A(16xKC) and B(128xKC) tiles into padded LDS,
// WMMA frags come from ds_load_b64.
// epi: 0 = +bias(if non-null), 1 = relu(c+bias), 2 = c+bias+res
// ---------------------------------------------------------------------------
__global__ __launch_bounds__(256) void gemm_wmma_tdm(
    const float* __restrict__ A, const float* __restrict__ Bw,
    float* __restrict__ C, const float* __restrict__ bias,
    const float* __restrict__ res, int M, int K, int OUT, int epi)
{
  extern __shared__ float smem[];                 // [128*66] B tile | [16*66] A tile
  float* sB = smem;
  float* sA = smem + 128 * LDS_ROWSTRIDE;

  const int lane = threadIdx.x & 31;
  const int wave = threadIdx.x >> 5;
  const int half = lane >> 4;
  const int l    = lane & 15;
  const int rowBase = blockIdx.x * 16;
  const int colBase = blockIdx.y * 128;           // 8 waves cover 128 cols
  if (rowBase >= M) return;                       // block-uniform
  const int rowsLeft = (M - rowBase) < 16 ? (M - rowBase) : 16;

  const unsigned ldsB = (unsigned)(uintptr_t)sB;  // low 32 bits = LDS byte addr
  const unsigned ldsA = (unsigned)(uintptr_t)sA;

  v8f c = {0.f, 0.f, 0.f, 0.f, 0.f, 0.f, 0.f, 0.f};

  for (int k0 = 0; k0 < K; k0 += KCHUNK) {
    __syncthreads();                              // prior-chunk LDS reads done
    if (wave == 0) {
      tdm_load_tile_2d(ldsB, Bw + (size_t)colBase * K + k0, 128, K, KCHUNK, 128);
      tdm_load_tile_2d(ldsA, A  + (size_t)rowBase * K + k0, rowsLeft, K, KCHUNK, 16);
      __builtin_amdgcn_s_wait_tensorcnt(0);
    }
    __syncthreads();                              // data visible to all waves

    const float* ap = sA + l * LDS_ROWSTRIDE + 2 * half;
    const float* bp = sB + (wave * 16 + l) * LDS_ROWSTRIDE + 2 * half;
#pragma unroll
    for (int kk = 0; kk < KCHUNK; kk += 4) {
      // A 16x4 f32 frag: VGPR0 = {K0 | K2}, VGPR1 = {K1 | K3} (ISA 7.12.2)
      v2f a = *(const v2f*)(ap + kk);
      v2f b = *(const v2f*)(bp + kk);
      c = __builtin_amdgcn_wmma_f32_16x16x4_f32(false, a, false, b,
                                                (short)0, c, false, false);
    }
  }

  const int col = colBase + wave * 16 + l;
  const float bv = bias ? bias[col] : 0.f;
#pragma unroll
  for (int r = 0; r < 8; ++r) {
    const int row = rowBase + r + 8 * half;       // C: VGPR r -> M=r / M=r+8
    if (row < M) {
      float v = c[r] + bv;
      if (epi == 1)      v = fmaxf(v, 0.f);
      else if (epi == 2) v += res[(size_t)row * OUT + col];
      C[(size_t)row * OUT + col] = v;
    }
  }
}

// ---------------------------------------------------------------------------
// agg[n, d] = b[d]   (segment-sum bias init)
// ---------------------------------------------------------------------------
__global__ __launch_bounds__(256) void init_bias(float* __restrict__ agg,
                                                 const float* __restrict__ b,
                                                 int total)
{
  int i = blockIdx.x * 256 + threadIdx.x;
  if (i < total) agg[i] = b[i & (D_DIM - 1)];
}

// ---------------------------------------------------------------------------
// One wave per edge: msg = relu(h[src] + edge_attr[e]); atomic scatter to agg.
// edge_attr streams once -> non-temporal loads keep h/agg resident in L2.
// Prefetch the h row of edge e+2048 to hide gather latency.
// ---------------------------------------------------------------------------
__global__ __launch_bounds__(256) void edge_scatter(
    const float* __restrict__ h, const float* __restrict__ ea,
    const int* __restrict__ src, const int* __restrict__ dst,
    float* __restrict__ agg, int E)
{
  const int e = (blockIdx.x * 256 + threadIdx.x) >> 5;
  const int lane = threadIdx.x & 31;
  if (e >= E) return;

  const int PF = 2048;                            // ~wave-batch lookahead
  if (e + PF < E) {
    int sn = src[e + PF];
    __builtin_prefetch(h + (size_t)sn * D_DIM + lane * 4, 0, 1);
  }

  const int s = src[e];
  const int d = dst[e];
  const v4f hv = *(const v4f*)(h + (size_t)s * D_DIM + lane * 4);
  const v4f ev = __builtin_nontemporal_load(
      (const v4f*)(ea + (size_t)e * D_DIM + lane * 4));
  float m0 = fmaxf(hv[0] + ev[0], 0.f);
  float m1 = fmaxf(hv[1] + ev[1], 0.f);
  float m2 = fmaxf(hv[2] + ev[2], 0.f);
  float m3 = fmaxf(hv[3] + ev[3], 0.f);
  float* out = agg + (size_t)d * D_DIM + lane * 4;
  atomicAdd(out + 0, m0);
  atomicAdd(out + 1, m1);
  atomicAdd(out + 2, m2);
  atomicAdd(out + 3, m3);
}

// ---------------------------------------------------------------------------
// Column stats stage 1: per-block partial sum / sumsq (coalesced, 2 rows/iter)
// ---------------------------------------------------------------------------
__global__ __launch_bounds__(256) void col_stats_partial(
    const float* __restrict__ X, float* __restrict__ ps,
    float* __restrict__ pq, int nrows)
{
  const int c = threadIdx.x & (D_DIM - 1);
  const int rh = threadIdx.x >> 7;
  float s = 0.f, q = 0.f;
  for (int r = blockIdx.x * 2 + rh; r < nrows; r += gridDim.x * 2) {
    float v = X[(size_t)r * D_DIM + c];
    s += v; q += v * v;
  }
  __shared__ float ls[256], lq[256];
  ls[threadIdx.x] = s; lq[threadIdx.x] = q;
  __syncthreads();
  if (threadIdx.x < D_DIM) {
    ps[blockIdx.x * D_DIM + threadIdx.x] = ls[threadIdx.x] + ls[threadIdx.x + D_DIM];
    pq[blockIdx.x * D_DIM + threadIdx.x] = lq[threadIdx.x] + lq[threadIdx.x + D_DIM];
  }
}

// ---------------------------------------------------------------------------
// Stage 2 (deterministic, sequential over partials):
// ss[c] = g*invstd ; ss[128+c] = beta - mean*g*invstd
// ---------------------------------------------------------------------------
__global__ __launch_bounds__(128) void col_stats_final(
    const float* __restrict__ ps, const float* __restrict__ pq,
    int nblocks, float invN, const float* __restrict__ g,
    const float* __restrict__ beta, float* __restrict__ ss)
{
  const int c = threadIdx.x;
  float s = 0.f, q = 0.f;
  for (int i = 0; i < nblocks; ++i) {
    s += ps[i * D_DIM + c];
    q += pq[i * D_DIM + c];
  }
  const float m   = s * invN;
  const float var = q * invN - m * m;
  const float sc  = g[c] * rsqrtf(var + EPSBN);
  ss[c]          = sc;
  ss[D_DIM + c]  = beta[c] - m * sc;
}

// ---------------------------------------------------------------------------
// y = x + relu(agg*scale + shift), fused with partial stats of y
// (row-strided like col_stats_partial -> deterministic partials, saves a pass)
// ---------------------------------------------------------------------------
__global__ __launch_bounds__(256) void bn_relu_residual_stats(
    const float* __restrict__ agg, const float* __restrict__ x,
    const float* __restrict__ ss, float* __restrict__ y,
    float* __restrict__ ps, float* __restrict__ pq, int nrows)
{
  const int c = threadIdx.x & (D_DIM - 1);
  const int rh = threadIdx.x >> 7;
  const float sc = ss[c], sh = ss[D_DIM + c];
  float s = 0.f, q = 0.f;
  for (int r = blockIdx.x * 2 + rh; r < nrows; r += gridDim.x * 2) {
    const size_t i = (size_t)r * D_DIM + c;
    const float v = x[i] + fmaxf(agg[i] * sc + sh, 0.f);
    y[i] = v;
    s += v; q += v * v;
  }
  __shared__ float ls[256], lq[256];
  ls[threadIdx.x] = s; lq[threadIdx.x] = q;
  __syncthreads();
  if (threadIdx.x < D_DIM) {
    ps[blockIdx.x * D_DIM + threadIdx.x] = ls[threadIdx.x] + ls[threadIdx.x + D_DIM];
    pq[blockIdx.x * D_DIM + threadIdx.x] = lq[threadIdx.x] + lq[threadIdx.x + D_DIM];
  }
}

// out[i] = in[i]*scale + shift
__global__ __launch_bounds__(256) void bn_apply(
    const float* __restrict__ in, const float* __restrict__ ss,
    float* __restrict__ out, int total)
{
  int i = blockIdx.x * 256 + threadIdx.x;
  if (i < total) {
    int c = i & (D_DIM - 1);
    out[i] = in[i] * ss[c] + ss[D_DIM + c];
  }
}

// ---------------------------------------------------------------------------
extern "C" void kernel_launch(void* const* d_in, const int* in_sizes, int n_in,
                              void* d_out, int out_size, void* d_ws, size_t ws_size,
                              hipStream_t stream)
{
  const float* x     = (const float*)d_in[0];
  const float* ea    = (const float*)d_in[1];
  const float* W     = (const float*)d_in[2];
  const float* b     = (const float*)d_in[3];
  const float* bn_g  = (const float*)d_in[4];
  const float* bn_b  = (const float*)d_in[5];
  const float* bnl_g = (const float*)d_in[6];
  const float* bnl_b = (const float*)d_in[7];
  const float* bn2_g = (const float*)d_in[8];
  const float* bn2_b = (const float*)d_in[9];
  const float* W1    = (const float*)d_in[10];
  const float* b1    = (const float*)d_in[11];
  const float* W2    = (const float*)d_in[12];
  const float* b2    = (const float*)d_in[13];
  const int*   ei    = (const int*)d_in[14];

  const int N = in_sizes[0] / D_DIM;        // 100000
  const int E = in_sizes[14] / 2;           // 1600000
  const int F = in_sizes[11];               // 256
  const size_t ND = (size_t)N * D_DIM;
  const size_t NF = (size_t)N * F;

  // Workspace: B0=h->u [N*D] | B1=agg->yl [N*D] | B2=y1->t [N*F] | PS | PQ | SS
  float* ws = (float*)d_ws;
  float* B0 = ws;
  float* B1 = B0 + ND;
  float* B2 = B1 + ND;
  float* PS = B2 + NF;
  float* PQ = PS + STAT_BLOCKS * D_DIM;
  float* SS0 = PQ + STAT_BLOCKS * D_DIM;
  float* SS1 = SS0 + 2 * D_DIM;
  float* SS2 = SS1 + 2 * D_DIM;

  const int rowTiles = (N + 15) / 16;
  const int ewBlocks = ((E << 5) + 255) / 256;       // 1 wave per edge
  const int ndBlocks = (int)((ND + 255) / 256);
  const float invN = 1.0f / (float)N;
  const size_t smemB = (size_t)(128 + 16) * LDS_ROWSTRIDE * 4 + 16; // + pad slack

  // 1) h = x @ W^T
  gemm_wmma_tdm<<<dim3(rowTiles, D_DIM / 128), 256, smemB, stream>>>(
      x, W, B0, nullptr, nullptr, N, D_DIM, D_DIM, 0);

  // 2) agg = b broadcast, 3) edge message + scatter-add (L2-resident atomics)
  init_bias<<<ndBlocks, 256, 0, stream>>>(B1, b, (int)ND);
  edge_scatter<<<ewBlocks, 256, 0, stream>>>(B0, ea, ei, ei + E, B1, E);

  // 4) BN(agg) stats -> SS0
  col_stats_partial<<<STAT_BLOCKS, 256, 0, stream>>>(B1, PS, PQ, N);
  col_stats_final<<<1, 128, 0, stream>>>(PS, PQ, STAT_BLOCKS, invN, bn_g, bn_b, SS0);

  // 5) y1 = x + relu(bn(agg)), fused with y1 stats -> SS1
  bn_relu_residual_stats<<<STAT_BLOCKS, 256, 0, stream>>>(B1, x, SS0, B2, PS, PQ, N);
  col_stats_final<<<1, 128, 0, stream>>>(PS, PQ, STAT_BLOCKS, invN, bnl_g, bnl_b, SS1);

  // 6) yl = bn_l(y1) -> B1
  bn_apply<<<ndBlocks, 256, 0, stream>>>(B2, SS1, B1, (int)ND);

  // 7) t = relu(yl @ W1^T + b1)   [N, F]
  gemm_wmma_tdm<<<dim3(rowTiles, F / 128), 256, smemB, stream>>>(
      B1, W1, B2, b1, nullptr, N, D_DIM, F, 1);

  // 8) u = yl + t @ W2^T + b2     [N, D]
  gemm_wmma_tdm<<<dim3(rowTiles, D_DIM / 128), 256, smemB, stream>>>(
      B2, W2, B0, b2, B1, N, F, D_DIM, 2);

  // 9) BN2 stats -> SS2, out = bn2(u)
  col_stats_partial<<<STAT_BLOCKS, 256, 0, stream>>>(B0, PS, PQ, N);
  col_stats_final<<<1, 128, 0, stream>>>(PS, PQ, STAT_BLOCKS, invN, bn2_g, bn2_b, SS2);
  bn_apply<<<ndBlocks, 256, 0, stream>>>(B0, SS2, (float*)d_out, (int)ND);
}